// DMP_70952859730581
// MI455X (gfx1250) — compile-verified
//
#include <hip/hip_runtime.h>

#define DIM     1024
#define NB      25
#define KPAD    28          // K padded to multiple of 4 for 16x16x4 WMMA
#define TSTEPS  10000       // T-1 scan steps
#define ROWF    3072        // 3*DIM output row
#define LCH     100         // chunk length
#define NCH     100         // number of chunks (LCH*NCH == TSTEPS)

typedef __attribute__((ext_vector_type(2))) float v2f;
typedef __attribute__((ext_vector_type(4))) float v4f;
typedef __attribute__((ext_vector_type(8))) float v8f;

// ---- compile-time A^LCH (2x2 companion matrix of the integrator), in double ----
struct M2d { double a, b, c, d; };
constexpr M2d m2mul(M2d x, M2d y) {
    return { x.a*y.a + x.b*y.c, x.a*y.b + x.b*y.d,
             x.c*y.a + x.d*y.c, x.c*y.b + x.d*y.d };
}
constexpr M2d m2pow(M2d x, int n) {
    M2d r{1.0, 0.0, 0.0, 1.0};
    for (int i = 0; i < n; ++i) r = m2mul(r, x);
    return r;
}
constexpr double KTD_D = 0.001;                  // TAU*DT
constexpr double A21_D = -KTD_D * 25.0 * 6.25;   // -k*alpha_y*beta_y = -0.15625
constexpr double A22_D = 1.0 - KTD_D * 25.0;     // 1 - k*alpha_y     =  0.975
// state s=[y;dy]:  s' = A2*s + [k*beta; beta]
constexpr M2d A2  { 1.0 + KTD_D*A21_D, KTD_D*A22_D, A21_D, A22_D };
constexpr M2d ALM = m2pow(A2, LCH);
constexpr float AL00 = (float)ALM.a, AL01 = (float)ALM.b;
constexpr float AL10 = (float)ALM.c, AL11 = (float)ALM.d;

__device__ __forceinline__ void dmp_step(float f, float abg, float& y, float& dy) {
    float beta = 0.001f * (abg + f);
    float ndy  = fmaf(-0.15625f, y, fmaf(0.975f, dy, beta));
    y  = fmaf(0.001f, ndy, y);
    dy = ndy;
}

// ---------- phase 1: scaled psi rows  PsiS[t][j] = psi_j(x_t) * 1000*x_t/sum(psi) ----------
__global__ void k_psi(float* __restrict__ psis) {
    int t = blockIdx.x * blockDim.x + threadIdx.x;
    if (t >= TSTEPS) return;
    const float logdecay = -1.0005003335835335e-3f;       // ln(0.999)
    float x = expf((float)(t + 1) * logdecay);            // x after (t+1) decays
    float p[NB];
    float sum = 0.f;
    #pragma unroll
    for (int j = 0; j < NB; ++j) {
        float cj = expf(-(10.0f / 24.0f) * (float)j);     // exp(-ALPHA_X*linspace)
        float hj = 25.0f / cj;
        float dx = x - cj;
        float pj = expf(-hj * dx * dx);
        p[j] = pj; sum += pj;
    }
    float sc = 1000.0f * x / sum;
    float* row = psis + t * KPAD;
    #pragma unroll
    for (int j = 0; j < NB; ++j) row[j] = p[j] * sc;
    row[25] = 0.f; row[26] = 0.f; row[27] = 0.f;
}

// ---------- phase 1b: padded, (g-y0)-scaled weights  Wp[d][j] = W[d][j]*(g[d]-y0[d]) ----------
__global__ void k_wp(const float* __restrict__ w, const float* __restrict__ g,
                     const float* __restrict__ y0, float* __restrict__ wp) {
    int d = blockIdx.x * blockDim.x + threadIdx.x;
    if (d >= DIM) return;
    float s = g[d] - y0[d];
    const float* src = w + d * NB;
    float* dst = wp + d * KPAD;
    #pragma unroll
    for (int j = 0; j < NB; ++j) dst[j] = src[j] * s;
    dst[25] = 0.f; dst[26] = 0.f; dst[27] = 0.f;
}

// ---------- phase 2: F[t][d] = Wp @ PsiS^T via V_WMMA_F32_16X16X4_F32 ----------
// 64 M-tiles (dims) x 625 N-tiles (time) = 40000 waves, 7 wmma per tile (K=28).
__global__ void k_gemm(const float* __restrict__ wp, const float* __restrict__ psis,
                       float* __restrict__ F) {
    int wid  = (blockIdx.x * blockDim.x + threadIdx.x) >> 5;
    int lane = threadIdx.x & 31;
    int mt = wid & 63;      // dim tile
    int nt = wid >> 6;      // time tile
    if (nt >= TSTEPS / 16) return;
    int row = lane & 15;    // M for A, N for B (ISA 7.12.2 f32 layouts)
    int hi  = lane >> 4;    // selects K pair {0,1} vs {2,3} within each K-block of 4
    const float* arow = wp   + (size_t)(mt * 16 + row) * KPAD + hi * 2;
    const float* brow = psis + (size_t)(nt * 16 + row) * KPAD + hi * 2;
    v8f c = {};
    #pragma unroll
    for (int kb = 0; kb < KPAD / 4; ++kb) {
        v2f a = *(const v2f*)(arow + kb * 4);
        v2f b = *(const v2f*)(brow + kb * 4);
        c = __builtin_amdgcn_wmma_f32_16x16x4_f32(false, a, false, b,
                                                  (short)0, c, false, false);
    }
    // lane holds D[M=hi*8+r][N=row], r=0..7 -> 8 consecutive dims: two b128 stores
    float* dst = F + (size_t)(nt * 16 + row) * DIM + mt * 16 + hi * 8;
    v4f lo, hi4;
    lo.x  = c[0]; lo.y  = c[1]; lo.z  = c[2]; lo.w  = c[3];
    hi4.x = c[4]; hi4.y = c[5]; hi4.z = c[6]; hi4.w = c[7];
    *(v4f*)(dst)     = lo;
    *(v4f*)(dst + 4) = hi4;
}

// ---------- phase 3a: per-chunk partial (zero initial state) ----------
__global__ void k_part_F(const float* __restrict__ F, const float* __restrict__ g,
                         float2* __restrict__ P) {
    int d  = blockIdx.x * blockDim.x + threadIdx.x;
    int ch = blockIdx.y;
    float abg = 156.25f * g[d];
    float y = 0.f, dy = 0.f;
    const float* Fp = F + (size_t)ch * LCH * DIM + d;
    for (int s = 0; s < LCH; ++s) {
        if (s + 12 < LCH) __builtin_prefetch(Fp + (size_t)(s + 12) * DIM, 0, 3);
        dmp_step(Fp[(size_t)s * DIM], abg, y, dy);
    }
    P[ch * DIM + d] = make_float2(y, dy);
}

__global__ void k_part_dot(const float* __restrict__ psis, const float* __restrict__ w,
                           const float* __restrict__ g, const float* __restrict__ y0,
                           float2* __restrict__ P) {
    __shared__ float sp[LCH * KPAD];
    int tid = threadIdx.x;
    int ch  = blockIdx.y;
    for (int i = tid; i < LCH * KPAD; i += blockDim.x)
        sp[i] = psis[(size_t)ch * LCH * KPAD + i];
    __syncthreads();
    int d = blockIdx.x * blockDim.x + tid;
    float gy = g[d] - y0[d];
    float wreg[NB];
    #pragma unroll
    for (int j = 0; j < NB; ++j) wreg[j] = w[d * NB + j] * gy;
    float abg = 156.25f * g[d];
    float y = 0.f, dy = 0.f;
    for (int s = 0; s < LCH; ++s) {
        float f = 0.f;
        #pragma unroll
        for (int j = 0; j < NB; ++j) f = fmaf(wreg[j], sp[s * KPAD + j], f);
        dmp_step(f, abg, y, dy);
    }
    P[ch * DIM + d] = make_float2(y, dy);
}

// ---------- phase 3b: prefix over chunk boundaries with compile-time A^LCH ----------
__global__ void k_prefix(const float* __restrict__ y0, const float2* __restrict__ P,
                         float2* __restrict__ S) {
    int d = threadIdx.x;                 // 1024 threads, one block
    float sy = y0[d], sdy = 0.f;
    for (int i = 0; i < NCH; ++i) {
        S[i * DIM + d] = make_float2(sy, sdy);
        float2 p = P[i * DIM + d];
        float ny  = fmaf(AL00, sy, fmaf(AL01, sdy, p.x));
        float ndy = fmaf(AL10, sy, fmaf(AL11, sdy, p.y));
        sy = ny; sdy = ndy;
    }
}

// ---------- phase 3c: replay chunks from exact start states, stream outputs ----------
__global__ void k_final_F(const float* __restrict__ F, const float* __restrict__ g,
                          const float* __restrict__ y0, const float2* __restrict__ S,
                          float* __restrict__ out) {
    int d  = blockIdx.x * blockDim.x + threadIdx.x;
    int ch = blockIdx.y;
    float abg = 156.25f * g[d];
    float y0d = y0[d];
    float2 s0 = S[ch * DIM + d];
    float y = s0.x, dy = s0.y;
    const float* Fp = F + (size_t)ch * LCH * DIM + d;
    float* op = out + ((size_t)ch * LCH + 1) * ROWF + d;
    for (int s = 0; s < LCH; ++s) {
        if (s + 12 < LCH) __builtin_prefetch(Fp + (size_t)(s + 12) * DIM, 0, 3);
        dmp_step(Fp[(size_t)s * DIM], abg, y, dy);
        op[0]       = y0d;
        op[DIM]     = y;
        op[2 * DIM] = dy;
        op += ROWF;
    }
}

__global__ void k_final_dot(const float* __restrict__ psis, const float* __restrict__ w,
                            const float* __restrict__ g, const float* __restrict__ y0,
                            const float2* __restrict__ S, float* __restrict__ out) {
    __shared__ float sp[LCH * KPAD];
    int tid = threadIdx.x;
    int ch  = blockIdx.y;
    for (int i = tid; i < LCH * KPAD; i += blockDim.x)
        sp[i] = psis[(size_t)ch * LCH * KPAD + i];
    __syncthreads();
    int d = blockIdx.x * blockDim.x + tid;
    float gy = g[d] - y0[d];
    float wreg[NB];
    #pragma unroll
    for (int j = 0; j < NB; ++j) wreg[j] = w[d * NB + j] * gy;
    float abg = 156.25f * g[d];
    float y0d = y0[d];
    float2 s0 = S[ch * DIM + d];
    float y = s0.x, dy = s0.y;
    float* op = out + ((size_t)ch * LCH + 1) * ROWF + d;
    for (int s = 0; s < LCH; ++s) {
        float f = 0.f;
        #pragma unroll
        for (int j = 0; j < NB; ++j) f = fmaf(wreg[j], sp[s * KPAD + j], f);
        dmp_step(f, abg, y, dy);
        op[0]       = y0d;
        op[DIM]     = y;
        op[2 * DIM] = dy;
        op += ROWF;
    }
}

// ---------- row 0: [y0 | y0 | 0] ----------
__global__ void k_row0(const float* __restrict__ y0, float* __restrict__ out) {
    int d = blockIdx.x * blockDim.x + threadIdx.x;
    if (d >= DIM) return;
    float v = y0[d];
    out[d] = v;
    out[DIM + d] = v;
    out[2 * DIM + d] = 0.f;
}

extern "C" void kernel_launch(void* const* d_in, const int* in_sizes, int n_in,
                              void* d_out, int out_size, void* d_ws, size_t ws_size,
                              hipStream_t stream) {
    (void)in_sizes; (void)n_in; (void)out_size;
    const float* y0 = (const float*)d_in[0];
    const float* g  = (const float*)d_in[1];
    const float* w  = (const float*)d_in[2];
    float* out = (float*)d_out;

    // workspace layout (256B aligned)
    size_t off = 0;
    auto take = [&](size_t nfloats) {
        size_t o = off;
        off += ((nfloats * sizeof(float)) + 255) & ~(size_t)255;
        return o;
    };
    size_t o_psis = take((size_t)TSTEPS * KPAD);
    size_t o_P    = take((size_t)NCH * DIM * 2);
    size_t o_S    = take((size_t)NCH * DIM * 2);
    size_t need_min = off;
    size_t o_wp   = take((size_t)DIM * KPAD);
    size_t o_F    = take((size_t)TSTEPS * DIM);
    size_t need_full = off;
    (void)need_min;

    char* ws = (char*)d_ws;
    float*  psis = (float*)(ws + o_psis);
    float2* P    = (float2*)(ws + o_P);
    float2* S    = (float2*)(ws + o_S);
    float*  wp   = (float*)(ws + o_wp);
    float*  F    = (float*)(ws + o_F);

    bool useF = (ws_size >= need_full);

    k_psi<<<(TSTEPS + 255) / 256, 256, 0, stream>>>(psis);

    if (useF) {
        k_wp<<<DIM / 256, 256, 0, stream>>>(w, g, y0, wp);
        // 64 * 625 tiles, 8 waves per 256-thread block -> 5000 blocks
        k_gemm<<<(64 * (TSTEPS / 16)) / 8, 256, 0, stream>>>(wp, psis, F);
        k_part_F<<<dim3(DIM / 256, NCH), 256, 0, stream>>>(F, g, P);
        k_prefix<<<1, DIM, 0, stream>>>(y0, P, S);
        k_final_F<<<dim3(DIM / 256, NCH), 256, 0, stream>>>(F, g, y0, S, out);
    } else {
        k_part_dot<<<dim3(DIM / 256, NCH), 256, 0, stream>>>(psis, w, g, y0, P);
        k_prefix<<<1, DIM, 0, stream>>>(y0, P, S);
        k_final_dot<<<dim3(DIM / 256, NCH), 256, 0, stream>>>(psis, w, g, y0, S, out);
    }

    k_row0<<<DIM / 256, 256, 0, stream>>>(y0, out);
}